// Attention_51840255263121
// MI455X (gfx1250) — compile-verified
//
#include <hip/hip_runtime.h>
#include <hip/hip_bf16.h>

// ---- CDNA5 wave32 WMMA helpers -------------------------------------------
typedef __attribute__((ext_vector_type(2))) float    v2f;
typedef __attribute__((ext_vector_type(8))) float    v8f;
typedef __attribute__((ext_vector_type(4))) unsigned v4u;
typedef __attribute__((ext_vector_type(8))) unsigned v8u;

__device__ __forceinline__ v8f wmma_f32_16x16x4(v2f a, v2f b, v8f c) {
    return __builtin_amdgcn_wmma_f32_16x16x4_f32(false, a, false, b, (short)0, c,
                                                 false, false);
}

// Problem constants
#define BATCH 2
#define SEQ   2048
#define CDIM  2048
#define HEADS 16
#define HDIM  128
#define BN    (BATCH * SEQ)     // 4096
#define QKVLD (3 * CDIM)        // 6144
#define LDP   130               // LDS row pitch in dwords (128 + 2 pad)

// ---------------------------------------------------------------------------
// TDM: stage a 16 x 128 fp32 tile (row stride QKVLD) from global into LDS.
// D# per CDNA5 ISA:
//   group0: count=1 | lds_addr | global_addr[56:0] | type=2
//   group1: data_size=4B, pad_enable, pad_interval=128dw, pad_amount=2dw,
//           tensor_dim0=6144, tensor_dim1=65536, tile=128x16,
//           tensor_dim0_stride=6144
// LDS result: 16 rows, pitch LDP=130 dwords (pad keeps ds_load_b64 B-frag
// reads on 16 distinct banks). Tracked with TENSORcnt.
// ---------------------------------------------------------------------------
__device__ __forceinline__ void tdm_load_tile_16x128(const float* src,
                                                     unsigned lds_off) {
    unsigned long long ga = (unsigned long long)(size_t)src;
    v4u g0;
    g0[0] = 1u;                                            // count=1, user mode
    g0[1] = lds_off;                                       // LDS byte address
    g0[2] = (unsigned)ga;                                  // global_addr[31:0]
    g0[3] = (unsigned)((ga >> 32) & 0x1FFFFFFull) | 0x80000000u; // hi | type=2
    v8u g1;
    g1[0] = (2u << 16)        // data_size = 4 bytes
          | (1u << 20)        // pad_enable
          | (6u << 22)        // pad_interval: 2^(6+1)=128 dwords
          | (1u << 25);       // pad_amount: 2 dwords
    g1[1] = (QKVLD & 0xFFFFu) << 16;            // tensor_dim0[15:0]
    g1[2] = (QKVLD >> 16) | (0u << 16);         // tensor_dim0[31:16]|dim1 lo
    g1[3] = 1u | (128u << 16);                  // dim1 hi (65536) | tile_dim0=128
    g1[4] = 16u;                                // tile_dim1=16, tile_dim2=0
    g1[5] = (unsigned)QKVLD;                    // tensor_dim0_stride[31:0]
    g1[6] = 0u;
    g1[7] = 0u;
    asm volatile("tensor_load_to_lds %0, %1" :: "s"(g0), "s"(g1) : "memory");
}

// ---------------------------------------------------------------------------
// Kernel 1/4: OUT[M x Nc] = A[M x K] * W[Nc x K]^T (+ bias), fp32 WMMA.
// Register-blocked: one wave computes a 32(M) x 64(N) tile -> per K-step of 4:
// 6 x b64 loads feed 8 WMMAs (0.75 loads/WMMA).
// ---------------------------------------------------------------------------
__global__ __launch_bounds__(256) void gemm_wmma_f32(
    const float* __restrict__ A, const float* __restrict__ W,
    const float* __restrict__ bias, float* __restrict__ out,
    int M, int Nc, int K)
{
    const int l    = threadIdx.x & 31;
    const int wave = threadIdx.x >> 5;
    const int ln   = l & 15;
    const int lg   = l >> 4;

    const int nBase = blockIdx.x * 64;
    const int mBase = (blockIdx.y * 8 + wave) * 32;
    if (mBase >= M || nBase >= Nc) return;

    const float* arow0 = A + (size_t)(mBase + ln) * K + lg * 2;
    const float* arow1 = arow0 + (size_t)16 * K;
    const float* wrow  = W + (size_t)(nBase + ln) * K + lg * 2;
    const size_t wstep = (size_t)16 * K;

    v8f acc[2][4];
#pragma unroll
    for (int mi = 0; mi < 2; ++mi)
#pragma unroll
        for (int t = 0; t < 4; ++t) acc[mi][t] = (v8f){};

#pragma unroll 4
    for (int k = 0; k < K; k += 4) {
        v2f a0 = *(const v2f*)(arow0 + k);
        v2f a1 = *(const v2f*)(arow1 + k);
        v2f b0 = *(const v2f*)(wrow + k);
        v2f b1 = *(const v2f*)(wrow + wstep + k);
        v2f b2 = *(const v2f*)(wrow + 2 * wstep + k);
        v2f b3 = *(const v2f*)(wrow + 3 * wstep + k);
        acc[0][0] = wmma_f32_16x16x4(a0, b0, acc[0][0]);
        acc[0][1] = wmma_f32_16x16x4(a0, b1, acc[0][1]);
        acc[0][2] = wmma_f32_16x16x4(a0, b2, acc[0][2]);
        acc[0][3] = wmma_f32_16x16x4(a0, b3, acc[0][3]);
        acc[1][0] = wmma_f32_16x16x4(a1, b0, acc[1][0]);
        acc[1][1] = wmma_f32_16x16x4(a1, b1, acc[1][1]);
        acc[1][2] = wmma_f32_16x16x4(a1, b2, acc[1][2]);
        acc[1][3] = wmma_f32_16x16x4(a1, b3, acc[1][3]);
    }

#pragma unroll
    for (int t = 0; t < 4; ++t) {
        const float bv = bias ? bias[nBase + t * 16 + ln] : 0.0f;
#pragma unroll
        for (int mi = 0; mi < 2; ++mi) {
#pragma unroll
            for (int r = 0; r < 8; ++r) {
                const int m = mBase + mi * 16 + r + 8 * lg;
                out[(size_t)m * Nc + nBase + t * 16 + ln] = acc[mi][t][r] + bv;
            }
        }
    }
}

// ---------------------------------------------------------------------------
// Kernel 2: fused RMSNorm + RoPE in-place on q (s=0) and k (s=1) rows.
// One wave per (bn, s, h) row of 128. Lane t owns d = {t, t+32} and the
// rotate_half partners {t+64, t+96}.
// ---------------------------------------------------------------------------
__global__ __launch_bounds__(256) void rmsnorm_rope(
    float* __restrict__ qkv,
    const float* __restrict__ rope_cos, const float* __restrict__ rope_sin,
    const float* __restrict__ q_norm_w, const float* __restrict__ k_norm_w)
{
    const int l    = threadIdx.x & 31;
    const int wave = threadIdx.x >> 5;
    int row = blockIdx.x * 8 + wave;              // [0, BN*2*HEADS)

    const int h  = row % HEADS;  row /= HEADS;
    const int s  = row & 1;      row >>= 1;
    const int bn = row;
    const int n  = bn & (SEQ - 1);

    float* p = qkv + (size_t)bn * QKVLD + s * CDIM + h * HDIM;
    const float* w  = s ? k_norm_w : q_norm_w;
    const float* cp = rope_cos + (size_t)n * HDIM;
    const float* sp = rope_sin + (size_t)n * HDIM;

    float x0 = p[l], x1 = p[l + 32], x2 = p[l + 64], x3 = p[l + 96];

    float ss = x0 * x0 + x1 * x1 + x2 * x2 + x3 * x3;
    ss += __shfl_xor(ss, 16);
    ss += __shfl_xor(ss, 8);
    ss += __shfl_xor(ss, 4);
    ss += __shfl_xor(ss, 2);
    ss += __shfl_xor(ss, 1);
    const float inv = rsqrtf(ss * (1.0f / 128.0f) + 1e-6f);

    x0 *= inv * w[l];
    x1 *= inv * w[l + 32];
    x2 *= inv * w[l + 64];
    x3 *= inv * w[l + 96];

    const float c0 = cp[l], c1 = cp[l + 32], c2 = cp[l + 64], c3 = cp[l + 96];
    const float s0 = sp[l], s1 = sp[l + 32], s2 = sp[l + 64], s3 = sp[l + 96];

    p[l]      = x0 * c0 - x2 * s0;
    p[l + 32] = x1 * c1 - x3 * s1;
    p[l + 64] = x2 * c2 + x0 * s2;
    p[l + 96] = x3 * c3 + x1 * s3;
}

// ---------------------------------------------------------------------------
// Kernel 3: flash attention, TDM-staged. All 8 waves of a block share one
// (b,h); wave 0 double-buffers the 16x128 K and V tiles into LDS with
// tensor_load_to_lds, waits on TENSORcnt, and the workgroup barrier releases
// the others. QK^T B-frags and P*V B-frags then come from LDS.
// ---------------------------------------------------------------------------
__global__ __launch_bounds__(256) void flash_attn(
    const float* __restrict__ qkv, float* __restrict__ out)
{
    __shared__ float kbuf[2][16 * LDP];
    __shared__ float vbuf[2][16 * LDP];
    __shared__ float plds[8][16 * 17];

    const int l    = threadIdx.x & 31;
    const int wave = threadIdx.x >> 5;
    const int ln   = l & 15;
    const int lg   = l >> 4;

    const int gw = blockIdx.x * 8 + wave;         // [0, 4096)
    const int qt = gw & 127;                      // query tile within (b,h)
    const int bh = gw >> 7;                       // shared by all waves in blk
    const int h  = bh & (HEADS - 1);
    const int b  = bh >> 4;

    const float* qbase = qkv + (size_t)b * SEQ * QKVLD + h * HDIM;
    const float* kbase = qbase + CDIM;
    const float* vbase = qbase + 2 * CDIM;
    float* myp = &plds[wave][0];

    // Cache Q fragments for the whole D=128 reduction (32 k-steps of 4).
    v2f qa[32];
    {
        const float* qrow = qbase + (size_t)(qt * 16 + ln) * QKVLD + lg * 2;
#pragma unroll
        for (int kb = 0; kb < 32; ++kb) qa[kb] = *(const v2f*)(qrow + kb * 4);
    }

    v8f acc[8];
#pragma unroll
    for (int i = 0; i < 8; ++i) acc[i] = (v8f){};
    float rm[8], rs[8];
#pragma unroll
    for (int r = 0; r < 8; ++r) { rm[r] = -3.0e38f; rs[r] = 0.0f; }

    const float scale = 0.08838834764831845f;     // 1/sqrt(128)

    // Prologue: stage tile j=0 into buffer 0.
    if (wave == 0) {
        tdm_load_tile_16x128(kbase, (unsigned)(size_t)&kbuf[0][0]);
        tdm_load_tile_16x128(vbase, (unsigned)(size_t)&vbuf[0][0]);
    }

    for (int j = 0; j < SEQ / 16; ++j) {
        const int buf = j & 1;
        if (wave == 0) __builtin_amdgcn_s_wait_tensorcnt(0);
        __syncthreads();    // buf[j&1] ready; all reads of buf[(j+1)&1] done

        if (wave == 0 && j + 1 < SEQ / 16) {
            const float* kn = kbase + (size_t)((j + 1) * 16) * QKVLD;
            const float* vn = vbase + (size_t)((j + 1) * 16) * QKVLD;
            tdm_load_tile_16x128(kn, (unsigned)(size_t)&kbuf[buf ^ 1][0]);
            tdm_load_tile_16x128(vn, (unsigned)(size_t)&vbuf[buf ^ 1][0]);
        }

        // ---- S = Q * K^T from LDS K tile -------------------------------
        v8f s = {};
        const float* kk = &kbuf[buf][ln * LDP + lg * 2];
#pragma unroll 8
        for (int kb = 0; kb < 32; ++kb) {
            v2f bf = *(const v2f*)(kk + kb * 4);
            s = wmma_f32_16x16x4(qa[kb], bf, s);
        }

        // ---- online softmax (row r+8*lg lives in element r) ------------
        float pr[8];
#pragma unroll
        for (int r = 0; r < 8; ++r) {
            float v = s[r] * scale;
            float mx = v;
            mx = fmaxf(mx, __shfl_xor(mx, 1));
            mx = fmaxf(mx, __shfl_xor(mx, 2));
            mx = fmaxf(mx, __shfl_xor(mx, 4));
            mx = fmaxf(mx, __shfl_xor(mx, 8));
            const float nm    = fmaxf(rm[r], mx);
            const float alpha = __expf(rm[r] - nm);
            rm[r] = nm;
            const float p = __expf(v - nm);
            pr[r] = p;
            float sum = p;
            sum += __shfl_xor(sum, 1);
            sum += __shfl_xor(sum, 2);
            sum += __shfl_xor(sum, 4);
            sum += __shfl_xor(sum, 8);
            rs[r] = rs[r] * alpha + sum;
#pragma unroll
            for (int t = 0; t < 8; ++t) acc[t][r] *= alpha;
        }

        // ---- C-layout -> A-layout via per-wave LDS tile ----------------
#pragma unroll
        for (int r = 0; r < 8; ++r) myp[(r + 8 * lg) * 17 + ln] = pr[r];
        __syncthreads();

        // ---- O += P * V from LDS V tile --------------------------------
#pragma unroll
        for (int dn = 0; dn < 8; ++dn) {
#pragma unroll
            for (int kb = 0; kb < 4; ++kb) {
                const int k0 = kb * 4 + lg * 2;
                v2f af = *(const v2f*)(myp + ln * 17 + k0);
                v2f bf;
                bf.x = vbuf[buf][k0 * LDP + dn * 16 + ln];
                bf.y = vbuf[buf][(k0 + 1) * LDP + dn * 16 + ln];
                acc[dn] = wmma_f32_16x16x4(af, bf, acc[dn]);
            }
        }
    }

    // ---- normalize and store: out is (B, N, H*D) row-major -------------
    float* obase = out + ((size_t)b * SEQ + qt * 16) * CDIM + h * HDIM;
#pragma unroll
    for (int r = 0; r < 8; ++r) {
        const float inv = 1.0f / rs[r];
        const int m = r + 8 * lg;
#pragma unroll
        for (int dn = 0; dn < 8; ++dn)
            obase[(size_t)m * CDIM + dn * 16 + ln] = acc[dn][r] * inv;
    }
}

// ---------------------------------------------------------------------------
extern "C" void kernel_launch(void* const* d_in, const int* in_sizes, int n_in,
                              void* d_out, int out_size, void* d_ws, size_t ws_size,
                              hipStream_t stream) {
    const float* x        = (const float*)d_in[0];
    const float* rope_cos = (const float*)d_in[1];
    const float* rope_sin = (const float*)d_in[2];
    const float* qkv_w    = (const float*)d_in[3];
    const float* proj_w   = (const float*)d_in[4];
    const float* proj_b   = (const float*)d_in[5];
    const float* q_norm_w = (const float*)d_in[6];
    const float* k_norm_w = (const float*)d_in[7];
    float* out = (float*)d_out;

    float* qkv  = (float*)d_ws;                       // BN x 6144
    float* attn = qkv + (size_t)BN * QKVLD;           // BN x 2048

    // 1) QKV projection: [4096 x 6144] = x[4096 x 2048] * qkv_w^T
    {
        dim3 grid(QKVLD / 64, BN / (32 * 8));
        gemm_wmma_f32<<<grid, 256, 0, stream>>>(x, qkv_w, nullptr, qkv,
                                                BN, QKVLD, CDIM);
    }
    // 2) RMSNorm + RoPE on q and k rows (in place)
    {
        const int rows = BN * 2 * HEADS;              // 131072
        rmsnorm_rope<<<rows / 8, 256, 0, stream>>>(qkv, rope_cos, rope_sin,
                                                   q_norm_w, k_norm_w);
    }
    // 3) Flash attention -> attn [4096 x 2048]
    {
        const int waves = BATCH * HEADS * (SEQ / 16); // 4096
        flash_attn<<<waves / 8, 256, 0, stream>>>(qkv, attn);
    }
    // 4) Output projection + bias -> d_out
    {
        dim3 grid(CDIM / 64, BN / (32 * 8));
        gemm_wmma_f32<<<grid, 256, 0, stream>>>(attn, proj_w, proj_b, out,
                                                BN, CDIM, CDIM);
    }
}